// ProposalProcessor_63299228009342
// MI455X (gfx1250) — compile-verified
//
#include <hip/hip_runtime.h>
#include <hip/hip_bf16.h>
#include <math.h>

// ---------------- problem constants (fixed by the reference) ----------------
constexpr int   IMG   = 48;            // H == W == 48
constexpr int   CH    = 5;             // channels per proposal
constexpr int   NB    = IMG * IMG;     // 2304 boxes per image
constexpr int   TPB   = 256;           // threads per block (8 wave32 waves)
constexpr int   PER   = NB / TPB;      // 9 boxes/ranks per thread
constexpr int   MS    = 4096;          // bitonic sort size (pow2 >= NB)
constexpr float NMS_THR   = 0.7f;
constexpr float SCORE_THR = 0.5f;

// ---------------- CDNA5 TDM descriptor vector types --------------------------
typedef unsigned int u32x4 __attribute__((ext_vector_type(4)));
typedef int          i32x8 __attribute__((ext_vector_type(8)));
typedef int          i32x4 __attribute__((ext_vector_type(4)));

struct SMem {
    union {
        float              tile[CH * NB];   // 46080 B : raw planes staged by TDM
        unsigned long long keys[MS];        // 32768 B : (score_bits<<32)|(~idx)
    } u;
    float bc[2][6];                         // double-buffered broadcast slot
};

__global__ __launch_bounds__(TPB)
void proposal_nms_kernel(const float* __restrict__ proposals,
                         const int*   __restrict__ img_h_p,
                         const int*   __restrict__ img_w_p,
                         float*       __restrict__ out, int B)
{
    __shared__ SMem sm;
    const int b = blockIdx.x;
    const int t = threadIdx.x;

    // ---------- Phase 0: TDM bulk load of this image's 5x2304 tile into LDS
    // One wave issues a single 2D tensor descriptor: tile_dim0 = 2304 elems
    // (4B), tile_dim1 = 5 rows, contiguous (tensor_dim0_stride = 2304).
    if (t == 0) {
        unsigned long long ga =
            (unsigned long long)(uintptr_t)(proposals + (size_t)b * (CH * NB));
        unsigned lds = (unsigned)(uintptr_t)(&sm.u.tile[0]); // LDS byte offset

        u32x4 g0;
        g0[0] = 1u;                                   // count = 1 valid D#
        g0[1] = lds;                                  // lds_addr
        g0[2] = (unsigned)(ga & 0xFFFFFFFFull);       // global_addr[31:0]
        g0[3] = (unsigned)((ga >> 32) & 0x1FFFFFFull) // global_addr[56:32]
              | (2u << 30);                           // type = 2 ("image")

        i32x8 g1;
        g1[0] = (int)(2u << 16);                      // data_size = 4 bytes
        g1[1] = (int)(((unsigned)NB & 0xFFFFu) << 16);        // tensor_dim0 lo16
        g1[2] = (int)(((unsigned)NB >> 16) | ((unsigned)CH << 16)); // dim0 hi | dim1 lo
        g1[3] = (int)(((unsigned)NB & 0xFFFFu) << 16);        // dim1 hi(0) | tile_dim0
        g1[4] = (int)CH;                              // tile_dim1 = 5, tile_dim2 = 0
        g1[5] = (int)NB;                              // tensor_dim0_stride lo32
        g1[6] = 0;                                    // stride hi | dim1_stride lo
        g1[7] = 0;

        i32x4 g2 = {0, 0, 0, 0};                      // 2D tensor: groups 2/3 unused
        i32x4 g3 = {0, 0, 0, 0};
        i32x8 g4 = {0, 0, 0, 0, 0, 0, 0, 0};          // 6-arg toolchain extra group
        __builtin_amdgcn_tensor_load_to_lds(g0, g1, g2, g3, g4, 0);
        __builtin_amdgcn_s_wait_tensorcnt(0);         // TENSORcnt == 0
    }
    __syncthreads();

    const float sx = (float)img_w_p[0] / (float)IMG;  // 768/48 = 16
    const float sy = (float)img_h_p[0] / (float)IMG;
    float4* outB    = reinterpret_cast<float4*>(out) + (size_t)b * NB;
    float*  keepOut = out + (size_t)B * NB * 4 + (size_t)b * NB;

    // ---------- Phase 1: decode; write UNSCALED boxes to d_out (rescaled later)
    float myScore[PER];
    #pragma unroll
    for (int k = 0; k < PER; ++k) {
        int n = t + k * TPB;
        int y = n / IMG, x = n - y * IMG;
        float p0 = sm.u.tile[0 * NB + n];
        float p1 = sm.u.tile[1 * NB + n];
        float p2 = sm.u.tile[2 * NB + n];
        float p3 = sm.u.tile[3 * NB + n];
        float p4 = sm.u.tile[4 * NB + n];
        float cx = (float)x + p0, cy = (float)y + p1;
        float wb = expf(p2), hb = expf(p3);
        outB[n] = make_float4(cx - 0.5f * wb, cy - 0.5f * hb,
                              cx + 0.5f * wb, cy + 0.5f * hb);
        myScore[k] = 1.0f / (1.0f + expf(-p4));       // sigmoid
    }
    __syncthreads();   // tile reads done (LDS reused below) + global stores fenced

    // ---------- Phase 2: build sort keys (stable descending order)
    // sigmoid > 0 always -> positive-float bits are monotone as u32.
    // secondary key ~idx -> ties resolved by smaller original index first.
    #pragma unroll
    for (int k = 0; k < PER; ++k) {
        int n = t + k * TPB;
        unsigned hi = __float_as_uint(myScore[k]);
        sm.u.keys[n] = ((unsigned long long)hi << 32)
                     | (unsigned long long)(0xFFFFFFFFu - (unsigned)n);
    }
    for (int e = NB + t; e < MS; e += TPB) sm.u.keys[e] = 0ull; // pad sinks to end
    __syncthreads();

    // ---------- Phase 3: bitonic sort, descending, 4096 keys in LDS
    for (int k = 2; k <= MS; k <<= 1) {
        for (int j = k >> 1; j > 0; j >>= 1) {
            for (int e = t; e < MS; e += TPB) {
                int x = e ^ j;
                if (x > e) {
                    unsigned long long a = sm.u.keys[e];
                    unsigned long long c = sm.u.keys[x];
                    bool desc = (e & k) == 0;                  // top level: all desc
                    if (desc ? (a < c) : (a > c)) {
                        sm.u.keys[e] = c; sm.u.keys[x] = a;
                    }
                }
            }
            __syncthreads();
        }
    }

    // ---------- Phase 4: gather sorted boxes into registers
    float    gx1[PER], gy1[PER], gx2[PER], gy2[PER], gar[PER];
    unsigned gidx[PER];
    unsigned validMask = 0u, suppMask = 0u;

    float top = __uint_as_float((unsigned)(sm.u.keys[0] >> 32));
    bool anyPass = top > SCORE_THR;    // rank 0 == jnp.argmax under stable order

    #pragma unroll
    for (int k = 0; k < PER; ++k) {
        int r = t + k * TPB;
        unsigned long long key = sm.u.keys[r];
        float    sc  = __uint_as_float((unsigned)(key >> 32));
        unsigned idx = 0xFFFFFFFFu - (unsigned)(key & 0xFFFFFFFFull);
        gidx[k] = idx;
        float4 bx = outB[idx];                        // unscaled box
        gx1[k] = bx.x; gy1[k] = bx.y; gx2[k] = bx.z; gy2[k] = bx.w;
        gar[k] = (bx.z - bx.x) * (bx.w - bx.y);
        bool v = anyPass ? (sc > SCORE_THR) : (r == 0);
        if (v) validMask |= (1u << k);
    }

    // ---------- Phase 5: greedy NMS, one barrier per step (double-buffered bcast)
    #pragma unroll 1
    for (int i = 0; i < NB; ++i) {
        int buf = i & 1;
        if (t == (i & (TPB - 1))) {
            int slot = i >> 8;                        // which of my PER ranks
            float kx1 = 0.f, ky1 = 0.f, kx2 = 0.f, ky2 = 0.f, ka = 0.f;
            unsigned kidx = 0u; bool vv = false, ss = false;
            #pragma unroll
            for (int s = 0; s < PER; ++s)             // compile-time select,
                if (s == slot) {                      // keeps arrays in VGPRs
                    kx1 = gx1[s]; ky1 = gy1[s]; kx2 = gx2[s]; ky2 = gy2[s];
                    ka = gar[s]; kidx = gidx[s];
                    vv = (validMask >> s) & 1u; ss = (suppMask >> s) & 1u;
                }
            bool keep = vv && !ss;
            keepOut[kidx] = keep ? 1.0f : 0.0f;       // decided exactly once
            sm.bc[buf][0] = keep ? 1.0f : 0.0f;
            sm.bc[buf][1] = kx1; sm.bc[buf][2] = ky1;
            sm.bc[buf][3] = kx2; sm.bc[buf][4] = ky2;
            sm.bc[buf][5] = ka;
        }
        __syncthreads();
        if (sm.bc[buf][0] != 0.0f) {                  // uniform branch
            float ix1 = sm.bc[buf][1], iy1 = sm.bc[buf][2];
            float ix2 = sm.bc[buf][3], iy2 = sm.bc[buf][4];
            float ia  = sm.bc[buf][5];
            #pragma unroll
            for (int k = 0; k < PER; ++k) {
                int r = t + k * TPB;
                if (r > i && !((suppMask >> k) & 1u)) {
                    float ltx = fmaxf(ix1, gx1[k]), lty = fmaxf(iy1, gy1[k]);
                    float rbx = fminf(ix2, gx2[k]), rby = fminf(iy2, gy2[k]);
                    float iw = fmaxf(rbx - ltx, 0.0f);
                    float ih = fmaxf(rby - lty, 0.0f);
                    float inter = iw * ih;
                    float iou = inter / (ia + gar[k] - inter);
                    if (iou > NMS_THR) suppMask |= (1u << k);
                }
            }
        }
    }

    // ---------- Phase 6: rescale boxes in place (all gathers are done)
    __syncthreads();
    #pragma unroll
    for (int k = 0; k < PER; ++k) {
        int n = t + k * TPB;
        float4 v = outB[n];
        v.x *= sx; v.y *= sy; v.z *= sx; v.w *= sy;
        outB[n] = v;
    }
}

extern "C" void kernel_launch(void* const* d_in, const int* in_sizes, int n_in,
                              void* d_out, int out_size, void* d_ws, size_t ws_size,
                              hipStream_t stream) {
    const float* proposals = (const float*)d_in[0];
    const int*   img_h     = (const int*)d_in[1];
    const int*   img_w     = (const int*)d_in[2];
    float*       out       = (float*)d_out;
    int B = in_sizes[0] / (CH * NB);                  // 16
    proposal_nms_kernel<<<dim3(B), dim3(TPB), 0, stream>>>(
        proposals, img_h, img_w, out, B);
}